// HGTModel_91053306675809
// MI455X (gfx1250) — compile-verified
//
#include <hip/hip_runtime.h>
#include <math.h>

typedef __attribute__((ext_vector_type(2))) float v2f;
typedef __attribute__((ext_vector_type(8))) float v8f;

#define NEG_SLOPE 0.2f

__device__ __forceinline__ float leaky(float v) { return v > 0.0f ? v : NEG_SLOPE * v; }

// ---------------------------------------------------------------------------
// WMMA f32 GEMM: C[M,N] = A[M,K] @ B[K,N] (+ bias per column, optional).
// KDIM/NDIM compile-time (64 or 128); M % 16 == 0 (true here: 50000, 800000).
//
// Block = 256 threads = 8 waves. Block stages all of B into LDS (<=64KB),
// interleaving the (k, k+1) element pair per column so each WMMA B-fragment
// is a single aligned ds_load_b64. Each wave owns one 16-row M-tile and all
// NDIM/16 N-tiles, so the A fragment (one global b64 per K-step) is reused
// across every WMMA of that step.
//
// Fragment layouts (V_WMMA_F32_16X16X4_F32):
//   A (16x4): lane = M + 16*(k>>1), vgpr = k&1
//   B (4x16): lane = N + 16*(k>>1), vgpr = k&1
//   C/D (16x16): lane L, vgpr r -> row = r + 8*(L>>4), col = L&15
// ---------------------------------------------------------------------------
template <int KDIM, int NDIM>
__global__ void gemm_wmma_f32_t(const float* __restrict__ A, const float* __restrict__ B,
                                const float* __restrict__ bias, float* __restrict__ C, int M) {
  constexpr int NT = NDIM / 16;  // N-tiles per wave (4 or 8)
  extern __shared__ float Bs[];  // KDIM*NDIM floats, pair-interleaved

  const int tid = threadIdx.x;
  // Cooperative fill: Bs[((k>>1)*NDIM + n)*2 + (k&1)] = B[k*NDIM + n]
  for (int i = tid; i < KDIM * NDIM; i += 256) {
    const int k = i / NDIM, n = i % NDIM;
    Bs[(((k >> 1) * NDIM + n) << 1) | (k & 1)] = B[i];
  }
  __syncthreads();

  const int wave = blockIdx.x * 8 + (tid >> 5);
  if (wave >= (M >> 4)) return;  // wave-uniform; after barrier, EXEC all-ones

  const int lane = tid & 31;
  const int half = lane >> 4;  // 0: lanes 0-15, 1: lanes 16-31
  const int l15  = lane & 15;
  const float* __restrict__ arow = A + (size_t)(wave * 16 + l15) * KDIM;

  v8f acc[NT];
#pragma unroll
  for (int t = 0; t < NT; ++t) acc[t] = {};

#pragma unroll
  for (int k = 0; k < KDIM; k += 4) {
    const int ka = k + 2 * half;                        // even
    const v2f a = *(const v2f*)(arow + ka);             // global b64
    const float* bp = Bs + (size_t)(ka >> 1) * (2 * NDIM);
#pragma unroll
    for (int t = 0; t < NT; ++t) {
      const v2f b = *(const v2f*)(bp + ((t * 16 + l15) << 1));  // ds_load_b64
      acc[t] = __builtin_amdgcn_wmma_f32_16x16x4_f32(
          /*neg_a=*/false, a, /*neg_b=*/false, b,
          /*c_mod=*/(short)0, acc[t], /*reuse_a=*/false, /*reuse_b=*/false);
    }
  }

#pragma unroll
  for (int t = 0; t < NT; ++t) {
    const int col = t * 16 + l15;
    const float bv = bias ? bias[col] : 0.0f;
    float* __restrict__ crow = C + (size_t)(wave * 16 + 8 * half) * NDIM + col;
#pragma unroll
    for (int r = 0; r < 8; ++r) crow[(size_t)r * NDIM] = acc[t][r] + bv;
  }
}

// ---------------------------------------------------------------------------
// Attention bookkeeping kernels (H = 2 heads, C = 64 per head hardcoded).
// xw rows are [n][128] = [n][h*64 + c].
// ---------------------------------------------------------------------------

// as_[n*2+h] = dot(xw[n][h*64:...], a_src[h]);  ad_ likewise.
__global__ void attn_coef(const float* __restrict__ xw, const float* __restrict__ a_src,
                          const float* __restrict__ a_dst, float* __restrict__ as_,
                          float* __restrict__ ad_, int n2 /* n*2 */) {
  int i = blockIdx.x * blockDim.x + threadIdx.x;
  if (i >= n2) return;
  const int node = i >> 1, h = i & 1;
  const float* row = xw + (size_t)node * 128 + h * 64;
  const float* vs = a_src + h * 64;
  const float* vd = a_dst + h * 64;
  float s = 0.0f, d = 0.0f;
  for (int c = 0; c < 64; ++c) { s += row[c] * vs[c]; d += row[c] * vd[c]; }
  as_[i] = s;
  ad_[i] = d;
}

// m[n*2+h] = leaky(as_[n][h] + ad_[n][h])   (self-loop seeds the segment max)
__global__ void init_m(const float* __restrict__ as_, const float* __restrict__ ad_,
                       float* __restrict__ m, int n2) {
  int i = blockIdx.x * blockDim.x + threadIdx.x;
  if (i >= n2) return;
  m[i] = leaky(as_[i] + ad_[i]);
}

// atomicMax over real edges: m[dst][h] = max(m[dst][h], leaky(as[src][h]+ad[dst][h]))
__global__ void edge_max(const int* __restrict__ src, const int* __restrict__ dst,
                         const float* __restrict__ as_, const float* __restrict__ ad_,
                         float* __restrict__ m, int e2 /* E*2 */) {
  int i = blockIdx.x * blockDim.x + threadIdx.x;
  if (i >= e2) return;
  const int e = i >> 1, h = i & 1;
  const int s = src[e], d = dst[e];
  atomicMax(&m[d * 2 + h], leaky(as_[s * 2 + h] + ad_[d * 2 + h]));
}

// den[n*2+h] = exp(self_e - m)   (self-loop seeds the segment sum)
__global__ void init_den(const float* __restrict__ as_, const float* __restrict__ ad_,
                         const float* __restrict__ m, float* __restrict__ den, int n2) {
  int i = blockIdx.x * blockDim.x + threadIdx.x;
  if (i >= n2) return;
  den[i] = expf(leaky(as_[i] + ad_[i]) - m[i]);
}

// den[dst][h] += exp(e_val - m[dst][h]) over real edges
__global__ void edge_den(const int* __restrict__ src, const int* __restrict__ dst,
                         const float* __restrict__ as_, const float* __restrict__ ad_,
                         const float* __restrict__ m, float* __restrict__ den, int e2) {
  int i = blockIdx.x * blockDim.x + threadIdx.x;
  if (i >= e2) return;
  const int e = i >> 1, h = i & 1;
  const int s = src[e], d = dst[e];
  const float v = leaky(as_[s * 2 + h] + ad_[d * 2 + h]);
  atomicAdd(&den[d * 2 + h], expf(v - m[d * 2 + h]));
}

// alpha[e*2+h] = exp(e_val - m[dst]) / den[dst]
__global__ void edge_alpha(const int* __restrict__ src, const int* __restrict__ dst,
                           const float* __restrict__ as_, const float* __restrict__ ad_,
                           const float* __restrict__ m, const float* __restrict__ den,
                           float* __restrict__ alpha, int e2) {
  int i = blockIdx.x * blockDim.x + threadIdx.x;
  if (i >= e2) return;
  const int e = i >> 1, h = i & 1;
  const int s = src[e], d = dst[e];
  const float v = leaky(as_[s * 2 + h] + ad_[d * 2 + h]);
  alpha[i] = expf(v - m[d * 2 + h]) / den[d * 2 + h];
}

// acc[n][idx] = alpha_self(n,h) * xw[n][idx]   (self-loop contribution)
__global__ void acc_init(const float* __restrict__ as_, const float* __restrict__ ad_,
                         const float* __restrict__ m, const float* __restrict__ den,
                         const float* __restrict__ xw, float* __restrict__ acc,
                         int n128 /* n*128 */) {
  int i = blockIdx.x * blockDim.x + threadIdx.x;
  if (i >= n128) return;
  const int node = i >> 7, idx = i & 127, h = idx >> 6;
  const float v = leaky(as_[node * 2 + h] + ad_[node * 2 + h]);
  const float a = expf(v - m[node * 2 + h]) / den[node * 2 + h];
  acc[i] = a * xw[i];
}

// acc[dst][idx] += alpha[e][h] * xw[src][idx]   (real edges, atomic scatter)
__global__ void edge_scatter(const int* __restrict__ src, const int* __restrict__ dst,
                             const float* __restrict__ alpha, const float* __restrict__ xw,
                             float* __restrict__ acc, int e128 /* E*128 */) {
  int i = blockIdx.x * blockDim.x + threadIdx.x;
  if (i >= e128) return;
  const int e = i >> 7, idx = i & 127, h = idx >> 6;
  const int s = src[e], d = dst[e];
  atomicAdd(&acc[(size_t)d * 128 + idx], alpha[e * 2 + h] * xw[(size_t)s * 128 + idx]);
}

// h_out[n][c] = 0.5*(acc[n][c] + acc[n][64+c]) + bias[c]    (head mean, H=2)
__global__ void head_mean(const float* __restrict__ acc, const float* __restrict__ bias,
                          float* __restrict__ hout, int n64 /* n*64 */) {
  int i = blockIdx.x * blockDim.x + threadIdx.x;
  if (i >= n64) return;
  const int node = i >> 6, c = i & 63;
  hout[i] = 0.5f * (acc[(size_t)node * 128 + c] + acc[(size_t)node * 128 + 64 + c]) + bias[c];
}

// ---------------------------------------------------------------------------
// Host-side driver (all launches on `stream`; kernels serialize in-order)
// ---------------------------------------------------------------------------
static inline int cdiv(int a, int b) { return (a + b - 1) / b; }

static void launch_gemm(const float* A, const float* B, const float* bias, float* C,
                        int M, int K, int N, hipStream_t stream) {
  const int blocks = cdiv(M >> 4, 8);
  const size_t lds = (size_t)K * N * sizeof(float);
  if (K == 128 && N == 128)
    gemm_wmma_f32_t<128, 128><<<blocks, 256, lds, stream>>>(A, B, bias, C, M);
  else if (K == 64 && N == 128)
    gemm_wmma_f32_t<64, 128><<<blocks, 256, lds, stream>>>(A, B, bias, C, M);
  else
    gemm_wmma_f32_t<64, 64><<<blocks, 256, lds, stream>>>(A, B, bias, C, M);
}

static void run_gat_phase(const float* xfeat, int nNodes, int Kin,
                          const float* W, const float* a_src, const float* a_dst,
                          const float* bconv, const float* Wdec, const float* bdec,
                          int outDim, const int* src, const int* dst, int nE,
                          float* ws, float* out, hipStream_t stream) {
  // workspace layout (floats)
  float* xw    = ws;                                  // nNodes*128
  float* acc   = xw  + (size_t)nNodes * 128;          // nNodes*128
  float* hfeat = acc + (size_t)nNodes * 128;          // nNodes*64
  float* as_   = hfeat + (size_t)nNodes * 64;         // nNodes*2
  float* ad_   = as_ + (size_t)nNodes * 2;            // nNodes*2
  float* m     = ad_ + (size_t)nNodes * 2;            // nNodes*2
  float* den   = m   + (size_t)nNodes * 2;            // nNodes*2
  float* alpha = den + (size_t)nNodes * 2;            // nE*2

  const int n2 = nNodes * 2, e2 = nE * 2;
  const int n128 = nNodes * 128, e128 = nE * 128, n64 = nNodes * 64;
  const int TPB = 256;

  // 1) xw = x @ W  -> [nNodes, 128]
  launch_gemm(xfeat, W, nullptr, xw, nNodes, Kin, 128, stream);
  // 2) per-node attention coefficients
  attn_coef<<<cdiv(n2, TPB), TPB, 0, stream>>>(xw, a_src, a_dst, as_, ad_, n2);
  // 3) segment max (self-loop seed, then atomicMax over edges)
  init_m<<<cdiv(n2, TPB), TPB, 0, stream>>>(as_, ad_, m, n2);
  edge_max<<<cdiv(e2, TPB), TPB, 0, stream>>>(src, dst, as_, ad_, m, e2);
  // 4) segment sum of exp
  init_den<<<cdiv(n2, TPB), TPB, 0, stream>>>(as_, ad_, m, den, n2);
  edge_den<<<cdiv(e2, TPB), TPB, 0, stream>>>(src, dst, as_, ad_, m, den, e2);
  // 5) per-edge softmax weight
  edge_alpha<<<cdiv(e2, TPB), TPB, 0, stream>>>(src, dst, as_, ad_, m, den, alpha, e2);
  // 6) weighted aggregation (self seed + atomic scatter)
  acc_init<<<cdiv(n128, TPB), TPB, 0, stream>>>(as_, ad_, m, den, xw, acc, n128);
  edge_scatter<<<cdiv(e128, TPB), TPB, 0, stream>>>(src, dst, alpha, xw, acc, e128);
  // 7) head mean + conv bias
  head_mean<<<cdiv(n64, TPB), TPB, 0, stream>>>(acc, bconv, hfeat, n64);
  // 8) decoder GEMM with fused bias -> out [nNodes, outDim]
  launch_gemm(hfeat, Wdec, bdec, out, nNodes, 64, outDim, stream);
}

extern "C" void kernel_launch(void* const* d_in, const int* in_sizes, int n_in,
                              void* d_out, int out_size, void* d_ws, size_t ws_size,
                              hipStream_t stream) {
  const float* x   = (const float*)d_in[0];   // [50000,128]
  const float* xe  = (const float*)d_in[1];   // [800000,64]
  const int*   ei  = (const int*)d_in[2];     // [2,800000]
  const float* Wn  = (const float*)d_in[3];   // [128,128]
  const float* asn = (const float*)d_in[4];   // [2,64]
  const float* adn = (const float*)d_in[5];
  const float* bn  = (const float*)d_in[6];   // [64]
  const float* We  = (const float*)d_in[7];   // [64,128]
  const float* ase = (const float*)d_in[8];
  const float* ade = (const float*)d_in[9];
  const float* be  = (const float*)d_in[10];
  const float* Wdx = (const float*)d_in[11];  // [64,128]
  const float* bdx = (const float*)d_in[12];  // [128]
  const float* Wde = (const float*)d_in[13];  // [64,64]
  const float* bde = (const float*)d_in[14];  // [64]

  const int N = 50000, E = 800000;
  const int* src = ei;
  const int* dst = ei + E;

  float* ws   = (float*)d_ws;
  float* xrec = (float*)d_out;                 // [N,128]
  float* erec = xrec + (size_t)N * 128;        // [E,64]

  // Node GAT phase (completes before edge phase on the same stream; workspace reused)
  run_gat_phase(x, N, 128, Wn, asn, adn, bn, Wdx, bdx, 128, src, dst, E, ws, xrec, stream);
  // Edge GAT phase: treats the E edge-feature rows as graph nodes (per reference)
  run_gat_phase(xe, E, 64, We, ase, ade, be, Wde, bde, 64, src, dst, E, ws, erec, stream);
}